// SSM_12936441496013
// MI455X (gfx1250) — compile-verified
//
#include <hip/hip_runtime.h>

typedef __attribute__((ext_vector_type(2))) float v2f;
typedef __attribute__((ext_vector_type(8))) float v8f;

#define H        1024          // hidden dim == scan block size
#define M_TILE   32            // rows per workgroup (2 row-tiles of 16)
#define K_BLOCK  64            // K slab staged in LDS per iteration
#define A_STRIDE (K_BLOCK + 4) // LDS pad (68 floats); multiple of 4 for float4
#define NTHREADS 512           // 16 waves
#define NWAVES   16

__global__ __launch_bounds__(NTHREADS)
void SSM_gemm_scan_kernel(const float* __restrict__ x,
                          const float* __restrict__ W,
                          const float* __restrict__ bias,
                          float* __restrict__ out)
{
    // 32 x 1024 fp32 output tile (128 KB) + A slab (8.5 KB) -> ~137 KB of 320 KB WGP LDS
    __shared__ float sOut[M_TILE * H];
    __shared__ float sA[M_TILE * A_STRIDE];

    const int tid  = threadIdx.x;
    const int wave = tid >> 5;       // 0..15
    const int lane = tid & 31;
    const int mBase = blockIdx.x * M_TILE;

    // Each wave: 64 output cols x 32 rows = 2 row-tiles x 4 col-tiles of 16x16
    // -> only 64 accumulator VGPRs, leaving room for deep B-fragment prefetch
    v8f acc[2][4];
    #pragma unroll
    for (int mt = 0; mt < 2; ++mt)
        #pragma unroll
        for (int nt = 0; nt < 4; ++nt)
            acc[mt][nt] = v8f{};

    const int nWaveBase = wave * 64;

    // A-fragment (16x4 f32): lanes 0-15 -> K=0,1 ; lanes 16-31 -> K=2,3 ; M = lane&15
    const int aRow  = lane & 15;
    const int aKoff = (lane >> 4) << 1;
    // B-fragment (4x16 f32): N = lane&15 ; same K split
    const int bLane = lane & 15;
    const int bKoff = (lane >> 4) << 1;

    for (int kb = 0; kb < H; kb += K_BLOCK) {
        // Stage x[mBase..+31][kb..+63]: 512 float4s, exactly one per thread
        {
            const int r  = tid >> 4;            // 0..31
            const int c4 = (tid & 15) << 2;     // 0,4,...,60
            const float4 v = *(const float4*)&x[(size_t)(mBase + r) * H + kb + c4];
            *(float4*)&sA[r * A_STRIDE + c4] = v;
        }
        __syncthreads();

        #pragma unroll
        for (int nt = 0; nt < 4; ++nt) {
            const int n = nWaveBase + nt * 16;
            const float* Wrow = W + (size_t)(n + bLane) * H + kb;  // W[o][k], o = n+bLane
            v8f a0 = acc[0][nt];
            v8f a1 = acc[1][nt];
            #pragma unroll
            for (int k = 0; k < K_BLOCK; k += 4) {
                // One B fragment feeds two row-tiles -> 1 global load : 2 wmmas
                v2f bfrag;
                bfrag.x = Wrow[k + bKoff + 0];
                bfrag.y = Wrow[k + bKoff + 1];
                v2f af0, af1;
                af0.x = sA[(aRow +  0) * A_STRIDE + k + aKoff + 0];
                af0.y = sA[(aRow +  0) * A_STRIDE + k + aKoff + 1];
                af1.x = sA[(aRow + 16) * A_STRIDE + k + aKoff + 0];
                af1.y = sA[(aRow + 16) * A_STRIDE + k + aKoff + 1];
                a0 = __builtin_amdgcn_wmma_f32_16x16x4_f32(
                        false, af0, false, bfrag, (short)0, a0, false, false);
                a1 = __builtin_amdgcn_wmma_f32_16x16x4_f32(
                        false, af1, false, bfrag, (short)0, a1, false, false);
            }
            acc[0][nt] = a0;
            acc[1][nt] = a1;
        }
        __syncthreads();
    }

    // Spill accumulators (+bias) into the LDS row tile.
    // C/D 16x16 f32 layout: VGPR j -> row j (lanes 0-15) / row j+8 (lanes 16-31), col = lane&15
    {
        const int cCol = lane & 15;
        const int rOff = (lane >> 4) << 3;   // 0 or 8
        #pragma unroll
        for (int nt = 0; nt < 4; ++nt) {
            const int n = nWaveBase + nt * 16 + cCol;
            const float b = bias[n];
            #pragma unroll
            for (int mt = 0; mt < 2; ++mt)
                #pragma unroll
                for (int j = 0; j < 8; ++j)
                    sOut[(mt * 16 + rOff + j) * H + n] = acc[mt][nt][j] + b;
        }
    }
    __syncthreads();

    // Per-row prefix sum over 1024 channels. 16 waves x 2 rows each.
    // Lane owns 32 contiguous elements; wave scan stitches lane partials.
    for (int r = wave; r < M_TILE; r += NWAVES) {
        const float* row = &sOut[r * H];
        const int base = lane * 32;

        float vals[32];
        float sum = 0.0f;
        #pragma unroll
        for (int i = 0; i < 32; i += 4) {
            const float4 v = *(const float4*)&row[base + i];
            vals[i + 0] = v.x; vals[i + 1] = v.y; vals[i + 2] = v.z; vals[i + 3] = v.w;
            sum += v.x + v.y + v.z + v.w;
        }

        // Inclusive wave32 scan of lane sums -> exclusive offset per lane
        float pre = sum;
        #pragma unroll
        for (int d = 1; d < 32; d <<= 1) {
            const float t = __shfl_up(pre, d, 32);
            if (lane >= d) pre += t;
        }
        float run = pre - sum;

        const size_t outBase = (size_t)(mBase + r) * H + base;
        #pragma unroll
        for (int i = 0; i < 32; i += 4) {
            float4 o;
            run += vals[i + 0]; o.x = run;
            run += vals[i + 1]; o.y = run;
            run += vals[i + 2]; o.z = run;
            run += vals[i + 3]; o.w = run;
            *(float4*)(out + outBase + i) = o;   // 16B aligned: base % 32 == 0
        }
    }
}

extern "C" void kernel_launch(void* const* d_in, const int* in_sizes, int n_in,
                              void* d_out, int out_size, void* d_ws, size_t ws_size,
                              hipStream_t stream) {
    const float* x    = (const float*)d_in[0];
    const float* W    = (const float*)d_in[1];
    const float* bias = (const float*)d_in[2];
    float* out = (float*)d_out;

    const int Mrows = in_sizes[0] / H;           // 8*4096 = 32768
    dim3 grid(Mrows / M_TILE);                   // 1024 workgroups
    SSM_gemm_scan_kernel<<<grid, NTHREADS, 0, stream>>>(x, W, bias, out);
}